// RNNDecoder_2327872275230
// MI455X (gfx1250) — compile-verified
//
#include <hip/hip_runtime.h>
#include <hip/hip_bf16.h>
#include <math.h>

// Problem constants (match reference)
#define V_ 32000
#define H_ 1024
#define B_ 32
#define S_ 64

typedef unsigned int       u32;
typedef unsigned long long u64;
typedef __attribute__((ext_vector_type(2))) float v2f;
typedef __attribute__((ext_vector_type(4))) float v4f;
typedef __attribute__((ext_vector_type(8))) float v8f;
typedef __attribute__((ext_vector_type(4))) u32   v4u;
typedef __attribute__((ext_vector_type(4))) int   v4i;
typedef __attribute__((ext_vector_type(8))) int   v8i;

#if __has_builtin(__builtin_amdgcn_tensor_load_to_lds) && \
    __has_builtin(__builtin_amdgcn_s_wait_tensorcnt)
#define USE_TDM 1
#else
#define USE_TDM 0
#endif

// ===========================================================================
// Kernel 1: small-M WMMA GEMM (GRU recurrence, M=32, N=3072).
// C[m,n] = sum_k Asrc[m,k] * Bm[n,k] + bias[n];  Asrc = relu(emb[token]) in
// gather mode.  BM=BN=64, BK=16, 256 threads = 8 waves, 16x32 per wave.
// ===========================================================================
#define BM 64
#define BN 64
#define BK 16
#define LDT 17

__global__ __launch_bounds__(256) void wmma_gemm_step_kernel(
    const float* __restrict__ A, const float* __restrict__ Bm,
    const float* __restrict__ bias, float* __restrict__ C,
    int M, int N, int K,
    const int* __restrict__ tgt, const int* __restrict__ bosp, int t,
    const float* __restrict__ emb)
{
    __shared__ float As[BM * LDT];
    __shared__ float Bs[BN * LDT];

    const int n0   = blockIdx.x * BN;
    const int m0   = blockIdx.y * BM;
    const int tid  = threadIdx.x;
    const int lane = tid & 31;
    const int wid  = tid >> 5;
    const int wm   = wid & 3;
    const int wn   = wid >> 2;

    v8f acc0 = {};
    v8f acc1 = {};

    const bool gather = (emb != nullptr);

    // Hoist per-row source pointers out of the K loop (token is K-invariant).
    const float* srcA[(BM * BK) / 256];
    const float* srcB[(BN * BK) / 256];
    #pragma unroll
    for (int i = 0; i < (BM * BK) / 256; ++i) {
        int idx = tid + i * 256;
        int r   = idx >> 4;
        int gm  = m0 + r;
        const float* p = nullptr;
        if (gm < M) {
            if (gather) {
                int tok = (t == 0) ? bosp[0] : tgt[gm * S_ + (t - 1)];
                p = emb + (long long)tok * K;
            } else {
                p = A + (long long)gm * K;
            }
        }
        srcA[i] = p;
    }
    #pragma unroll
    for (int i = 0; i < (BN * BK) / 256; ++i) {
        int idx = tid + i * 256;
        int r   = idx >> 4;
        int gn  = n0 + r;
        srcB[i] = (gn < N) ? (Bm + (long long)gn * K) : nullptr;
    }

    for (int k0 = 0; k0 < K; k0 += BK) {
        #pragma unroll
        for (int i = 0; i < (BM * BK) / 256; ++i) {
            int idx = tid + i * 256;
            int r   = idx >> 4;
            int kk  = idx & (BK - 1);
            float v = srcA[i] ? srcA[i][k0 + kk] : 0.0f;
            if (gather) v = v > 0.0f ? v : 0.0f;
            As[r * LDT + kk] = v;
        }
        #pragma unroll
        for (int i = 0; i < (BN * BK) / 256; ++i) {
            int idx = tid + i * 256;
            int r   = idx >> 4;
            int kk  = idx & (BK - 1);
            Bs[r * LDT + kk] = srcB[i] ? srcB[i][k0 + kk] : 0.0f;
        }
        __syncthreads();

        #pragma unroll
        for (int kk = 0; kk < BK; kk += 4) {
            const int koff = (lane >> 4) * 2;
            const int arow = wm * 16 + (lane & 15);
            v2f a;
            a.x = As[arow * LDT + kk + koff];
            a.y = As[arow * LDT + kk + koff + 1];

            const int brow = wn * 32 + (lane & 15);
            v2f b0, b1;
            b0.x = Bs[brow * LDT + kk + koff];
            b0.y = Bs[brow * LDT + kk + koff + 1];
            b1.x = Bs[(brow + 16) * LDT + kk + koff];
            b1.y = Bs[(brow + 16) * LDT + kk + koff + 1];

            acc0 = __builtin_amdgcn_wmma_f32_16x16x4_f32(
                false, a, false, b0, (short)0, acc0, false, false);
            acc1 = __builtin_amdgcn_wmma_f32_16x16x4_f32(
                false, a, false, b1, (short)0, acc1, false, false);
        }
        __syncthreads();
    }

    const int nbase    = n0 + wn * 32 + (lane & 15);
    const int mrow_off = (lane >> 4) * 8;
    #pragma unroll
    for (int r = 0; r < 8; ++r) {
        int gm = m0 + wm * 16 + r + mrow_off;
        if (gm < M) {
            int gn0 = nbase;
            int gn1 = nbase + 16;
            if (gn0 < N)
                C[(long long)gm * N + gn0] = acc0[r] + (bias ? bias[gn0] : 0.0f);
            if (gn1 < N)
                C[(long long)gm * N + gn1] = acc1[r] + (bias ? bias[gn1] : 0.0f);
        }
    }
}

// ===========================================================================
// Kernel 2: output projection GEMM (M=2048, N=32000, K=1024; tile-divisible).
// PBM=PBN=128, PBK=16.  256 threads = 8 waves; each wave owns a 32x64 strip
// = 2x4 fp32 16x16 WMMA accumulators.
// LDS staging via Tensor Data Mover: wave 0 issues one tensor_load_to_lds per
// matrix per K-chunk (double-buffered); the TDM pad feature (16 DWORDs data,
// 4 DWORDs pad) produces the bank-conflict-free stride of 20 floats.
// ===========================================================================
#define PBM 128
#define PBN 128
#define PBK 16
#define PLDT 20

#if USE_TDM
// D# group 0: count=1 | lds_addr | global_addr[56:0] | type=2
__device__ __forceinline__ v4u tdm_g0(u64 gaddr, u32 lds_byte)
{
    v4u g;
    g.x = 1u;                                  // count=1, user mode
    g.y = lds_byte;                            // lds_addr (bytes)
    g.z = (u32)(gaddr & 0xffffffffu);          // global_addr[31:0]
    g.w = (u32)((gaddr >> 32) & 0x01ffffffu)   // global_addr[56:32]
        | (2u << 30);                          // type = 2 ("image")
    return g;
}

// D# group 1: data_size=4B, pad_enable, pad_interval=16 DWORDs,
// pad_amount=4 DWORDs, tensor_dim0/1, tile_dim0/1, tensor_dim0_stride.
__device__ __forceinline__ v8i tdm_g1(u32 tdim0, u32 tdim1,
                                      u32 tile0, u32 tile1, u64 stride0)
{
    union { u64 q[4]; v8i v; } u;
    u.q[0] = (2ull << 16)                      // data_size: 4 bytes
           | (1ull << 20)                      // pad_enable
           | (3ull << 22)                      // pad_interval: 16 DWORDs
           | (3ull << 25)                      // pad_amount: 4 DWORDs
           | ((u64)(tdim0 & 0xffffu) << 48);   // tensor_dim0[15:0]
    u.q[1] = (u64)(tdim0 >> 16)                // tensor_dim0[31:16]
           | ((u64)tdim1 << 16)                // tensor_dim1
           | ((u64)(tile0 & 0xffffu) << 48);   // tile_dim0
    u.q[2] = (u64)(tile1 & 0xffffu)            // tile_dim1
           | ((stride0 & 0xffffffffull) << 32);// tensor_dim0_stride[31:0]
    u.q[3] = (stride0 >> 32) & 0xffffull;      // tensor_dim0_stride[47:32]
    return u.v;
}

// Generic LDS pointer -> LDS byte address (flat LDS aperture keeps the
// wave-relative offset in addr[31:0]).
__device__ __forceinline__ u32 lds_byte_addr(const void* p)
{
    return (u32)(u64)p;
}
#endif

__global__ __launch_bounds__(256) void wmma_gemm_proj_kernel(
    const float* __restrict__ A, const float* __restrict__ Bm,
    const float* __restrict__ bias, float* __restrict__ C,
    int M, int N, int K)
{
    __shared__ float As[2][PBM * PLDT];
    __shared__ float Bs[2][PBN * PLDT];

    const int n0   = blockIdx.x * PBN;
    const int m0   = blockIdx.y * PBM;
    const int tid  = threadIdx.x;
    const int lane = tid & 31;
    const int wid  = tid >> 5;
    const int wm   = wid & 3;    // row group of 32
    const int wn   = wid >> 2;   // col group of 64

    v8f acc[2][4] = {};

    const int nchunks = K / PBK;

#if USE_TDM
    const v4i zero4 = {0, 0, 0, 0};
    const v8i zero8 = {0, 0, 0, 0, 0, 0, 0, 0};
    const u64 gA0 = (u64)(A  + (long long)m0 * K);
    const u64 gB0 = (u64)(Bm + (long long)n0 * K);
    const v8i g1A = tdm_g1((u32)K, (u32)M, PBK, PBM, (u64)K);
    const v8i g1B = tdm_g1((u32)K, (u32)N, PBK, PBN, (u64)K);

    if (wid == 0) {
        // prologue: chunk 0 -> buffer 0
        __builtin_amdgcn_tensor_load_to_lds(
            tdm_g0(gA0, lds_byte_addr(&As[0][0])), g1A, zero4, zero4, zero8, 0);
        __builtin_amdgcn_tensor_load_to_lds(
            tdm_g0(gB0, lds_byte_addr(&Bs[0][0])), g1B, zero4, zero4, zero8, 0);
    }

    for (int i = 0; i < nchunks; ++i) {
        const int cur = i & 1;
        if (wid == 0) {
            if (i + 1 < nchunks) {
                const int nxt  = (i + 1) & 1;
                const u64 koff = (u64)(i + 1) * PBK * sizeof(float);
                __builtin_amdgcn_tensor_load_to_lds(
                    tdm_g0(gA0 + koff, lds_byte_addr(&As[nxt][0])),
                    g1A, zero4, zero4, zero8, 0);
                __builtin_amdgcn_tensor_load_to_lds(
                    tdm_g0(gB0 + koff, lds_byte_addr(&Bs[nxt][0])),
                    g1B, zero4, zero4, zero8, 0);
                __builtin_amdgcn_s_wait_tensorcnt(2); // chunk i complete
            } else {
                __builtin_amdgcn_s_wait_tensorcnt(0);
            }
        }
        __syncthreads();
#else
    // Fallback: synchronous VGPR-staged tiles (single buffer index 0).
    const int lrow = tid >> 1;
    const int lkk  = (tid & 1) * 8;
    const float* gArow = A  + (long long)(m0 + lrow) * K + lkk;
    const float* gBrow = Bm + (long long)(n0 + lrow) * K + lkk;
    float* sArow = &As[0][lrow * PLDT + lkk];
    float* sBrow = &Bs[0][lrow * PLDT + lkk];

    for (int i = 0; i < nchunks; ++i) {
        const int cur = 0;
        const int k0  = i * PBK;
        v4f a0 = *(const v4f*)(gArow + k0);
        v4f a1 = *(const v4f*)(gArow + k0 + 4);
        v4f b0 = *(const v4f*)(gBrow + k0);
        v4f b1 = *(const v4f*)(gBrow + k0 + 4);
        *(v4f*)(sArow)     = a0;
        *(v4f*)(sArow + 4) = a1;
        *(v4f*)(sBrow)     = b0;
        *(v4f*)(sBrow + 4) = b1;
        __syncthreads();
#endif

        #pragma unroll
        for (int kk = 0; kk < PBK; kk += 4) {
            const int koff2 = (lane >> 4) * 2;
            v2f af[2];
            #pragma unroll
            for (int mi = 0; mi < 2; ++mi) {
                const int arow = wm * 32 + mi * 16 + (lane & 15);
                af[mi].x = As[cur][arow * PLDT + kk + koff2];
                af[mi].y = As[cur][arow * PLDT + kk + koff2 + 1];
            }
            v2f bf[4];
            #pragma unroll
            for (int ni = 0; ni < 4; ++ni) {
                const int brow = wn * 64 + ni * 16 + (lane & 15);
                bf[ni].x = Bs[cur][brow * PLDT + kk + koff2];
                bf[ni].y = Bs[cur][brow * PLDT + kk + koff2 + 1];
            }
            #pragma unroll
            for (int mi = 0; mi < 2; ++mi)
                #pragma unroll
                for (int ni = 0; ni < 4; ++ni)
                    acc[mi][ni] = __builtin_amdgcn_wmma_f32_16x16x4_f32(
                        false, af[mi], false, bf[ni], (short)0, acc[mi][ni],
                        false, false);
        }
        __syncthreads();
    }

    // Epilogue: C/D layout => VGPR r is M row, lane half selects M+8.
    const int mrow_off = (lane >> 4) * 8;
    #pragma unroll
    for (int mi = 0; mi < 2; ++mi) {
        #pragma unroll
        for (int ni = 0; ni < 4; ++ni) {
            int gn = n0 + wn * 64 + ni * 16 + (lane & 15);
            float bv = bias ? bias[gn] : 0.0f;
            #pragma unroll
            for (int r = 0; r < 8; ++r) {
                int gm = m0 + wm * 32 + mi * 16 + r + mrow_off;
                C[(long long)gm * N + gn] = acc[mi][ni][r] + bv;
            }
        }
    }
}

// ---------------------------------------------------------------------------
// h = encoder_hidden[0]   (B*H copy)
// ---------------------------------------------------------------------------
__global__ __launch_bounds__(256) void init_h_kernel(
    const float* __restrict__ enc_hidden, float* __restrict__ h)
{
    int idx = blockIdx.x * 256 + threadIdx.x;
    if (idx < B_ * H_) h[idx] = enc_hidden[idx];
}

// ---------------------------------------------------------------------------
// GRU gate fusion: r,z,n gates + new hidden state.  hs laid out [B,S,H].
// ---------------------------------------------------------------------------
__global__ __launch_bounds__(256) void gru_gate_kernel(
    const float* __restrict__ gi, const float* __restrict__ gh,
    float* __restrict__ h, float* __restrict__ hs, int t)
{
    int idx = blockIdx.x * 256 + threadIdx.x;   // 0 .. B*H-1
    if (idx >= B_ * H_) return;
    int b = idx >> 10;
    int j = idx & (H_ - 1);

    const float* gib = gi + b * 3 * H_;
    const float* ghb = gh + b * 3 * H_;
    float ir = gib[j],          iz = gib[H_ + j],  in_ = gib[2 * H_ + j];
    float hr = ghb[j],          hz = ghb[H_ + j],  hn  = ghb[2 * H_ + j];

    float r = 1.0f / (1.0f + expf(-(ir + hr)));
    float z = 1.0f / (1.0f + expf(-(iz + hz)));
    float n = tanhf(in_ + r * hn);
    float hv = h[idx];
    float hnew = (1.0f - z) * n + z * hv;

    h[idx] = hnew;
    hs[((long long)b * S_ + t) * H_ + j] = hnew;
}

// ---------------------------------------------------------------------------
// In-place log_softmax over V, one block per row (3 passes, L2-resident row).
// ---------------------------------------------------------------------------
__global__ __launch_bounds__(256) void logsoftmax_kernel(float* __restrict__ out)
{
    long long row = blockIdx.x;
    float* p = out + row * (long long)V_;
    __shared__ float red[256];
    int tid = threadIdx.x;

    float m = -INFINITY;
    for (int i = tid; i < V_; i += 256) m = fmaxf(m, p[i]);
    red[tid] = m;
    __syncthreads();
    for (int s = 128; s > 0; s >>= 1) {
        if (tid < s) red[tid] = fmaxf(red[tid], red[tid + s]);
        __syncthreads();
    }
    m = red[0];
    __syncthreads();

    float sum = 0.0f;
    for (int i = tid; i < V_; i += 256) sum += expf(p[i] - m);
    red[tid] = sum;
    __syncthreads();
    for (int s = 128; s > 0; s >>= 1) {
        if (tid < s) red[tid] += red[tid + s];
        __syncthreads();
    }
    float lse = m + logf(red[0]);

    for (int i = tid; i < V_; i += 256) p[i] = p[i] - lse;
}

// ---------------------------------------------------------------------------
// hT output copy: out2[0][b][h] = h[b][h]
// ---------------------------------------------------------------------------
__global__ __launch_bounds__(256) void copy_ht_kernel(
    const float* __restrict__ h, float* __restrict__ dst)
{
    int idx = blockIdx.x * 256 + threadIdx.x;
    if (idx < B_ * H_) dst[idx] = h[idx];
}

// ---------------------------------------------------------------------------
// kernel_launch
// Inputs (setup_inputs order):
//  0: encoder_output (unused)  1: encoder_hidden  2: bos_token (int)
//  3: tgt_tensor (int)  4: emb  5: W_ih  6: W_hh  7: b_ih  8: b_hh
//  9: W_out  10: b_out
// Output: log_probs [B,S,V] then hT [1,B,H], flat fp32.
// ---------------------------------------------------------------------------
extern "C" void kernel_launch(void* const* d_in, const int* in_sizes, int n_in,
                              void* d_out, int out_size, void* d_ws, size_t ws_size,
                              hipStream_t stream)
{
    (void)in_sizes; (void)n_in; (void)out_size; (void)ws_size;

    const float* enc_hidden = (const float*)d_in[1];
    const int*   bosp       = (const int*)d_in[2];
    const int*   tgt        = (const int*)d_in[3];
    const float* emb        = (const float*)d_in[4];
    const float* W_ih       = (const float*)d_in[5];
    const float* W_hh       = (const float*)d_in[6];
    const float* b_ih       = (const float*)d_in[7];
    const float* b_hh       = (const float*)d_in[8];
    const float* W_out      = (const float*)d_in[9];
    const float* b_out      = (const float*)d_in[10];
    float*       out        = (float*)d_out;

    // Workspace layout (floats): h | gi | gh | hs   (~9.3 MB total)
    float* ws = (float*)d_ws;
    float* h  = ws;                              // B*H
    float* gi = h  + B_ * H_;                    // B*3H
    float* gh = gi + B_ * 3 * H_;                // B*3H
    float* hs = gh + B_ * 3 * H_;                // B*S*H

    dim3 blk(256);

    init_h_kernel<<<(B_ * H_ + 255) / 256, blk, 0, stream>>>(enc_hidden, h);

    // Sequential GRU recurrence
    dim3 gstep((3 * H_) / BN, (B_ + BM - 1) / BM);   // 48 x 1
    for (int t = 0; t < S_; ++t) {
        wmma_gemm_step_kernel<<<gstep, blk, 0, stream>>>(
            nullptr, W_ih, b_ih, gi, B_, 3 * H_, H_, tgt, bosp, t, emb);
        wmma_gemm_step_kernel<<<gstep, blk, 0, stream>>>(
            h, W_hh, b_hh, gh, B_, 3 * H_, H_, nullptr, nullptr, 0, nullptr);
        gru_gate_kernel<<<(B_ * H_) / 256, blk, 0, stream>>>(gi, gh, h, hs, t);
    }

    // logits = hs @ W_out^T + b_out  -> d_out  (M=2048, N=32000, K=1024)
    dim3 gproj(V_ / PBN, (B_ * S_) / PBM);           // 250 x 16
    wmma_gemm_proj_kernel<<<gproj, blk, 0, stream>>>(
        hs, W_out, b_out, out, B_ * S_, V_, H_);

    // in-place log_softmax over V, one block per (b,s) row
    logsoftmax_kernel<<<B_ * S_, blk, 0, stream>>>(out);

    // hT
    copy_ht_kernel<<<(B_ * H_ + 255) / 256, blk, 0, stream>>>(
        h, out + (long long)B_ * S_ * V_);
}